// ECE_41875931136678
// MI455X (gfx1250) — compile-verified
//
#include <hip/hip_runtime.h>

// ---------------------------------------------------------------------------
// Guided filter (color) for 1x3x2048x2048 f32, radius 16, eps 1e-4.
// Box filters implemented as banded-matrix GEMMs on the CDNA5 WMMA path
// (V_WMMA_F32_16X16X4_F32, f32 A/B, f32 accumulate -> numerically exact).
// Wave = 16-wide x 128-tall strip. Horizontal pass: band fragments hoisted,
// A loaded as b64 pairs with immediate offsets. Vertical pass: register ring
// reuses 32/48 rows per 16-row advance. All boundary handling is on
// wave-uniform branches around the load code only -> single WMMA loop,
// EXEC all-1s everywhere WMMA executes.
// ---------------------------------------------------------------------------

typedef __attribute__((ext_vector_type(2))) float v2f;
typedef __attribute__((ext_vector_type(8))) float v8f;

constexpr int IMW  = 2048;
constexpr int IMH  = 2048;
constexpr int IMHW = IMW * IMH;
constexpr int RAD  = 16;
constexpr int NCH1 = 21;   // guide(3) + src(3) + guide*src(9) + guide*guide(6)
constexpr int NCH2 = 12;   // a(9) + b(3)

__device__ __forceinline__ int clampi(int v, int lo, int hi) {
    return v < lo ? lo : (v > hi ? hi : v);
}

// ---------------------------------------------------------------------------
// Horizontal clamped box sum: out(16x16) = in(16x48) x Band(48x16).
// A layout (f32 16x4): lane&15 = M row; lanes 0-15 hold K pair {0,1},
// lanes 16-31 hold {2,3}.  B layout: lane&15 = N col; VGPRv holds row K.
// MODE 0: fused product channels from guide/src. MODE 1: plain plane buffer.
// ---------------------------------------------------------------------------
template <int MODE>
__global__ void hbox_kernel(const float* __restrict__ g, const float* __restrict__ s,
                            float* __restrict__ out) {
    const int lane = threadIdx.x & 31;
    const int wave = threadIdx.x >> 5;
    const int half = lane >> 4;          // K pair select
    const int mn   = lane & 15;          // M for A, N for B/D
    const int col0 = (blockIdx.x * 8 + wave) * 16;
    const int ch   = blockIdx.z;

    // Channel -> plane pointer(s), resolved once (scalar); replaces per-load switch.
    const float* __restrict__ p0;
    const float* __restrict__ p1 = nullptr;
    bool mul = false;
    if (MODE == 0) {
        if (ch < 3)       { p0 = g + (size_t)ch * IMHW; }
        else if (ch < 6)  { p0 = s + (size_t)(ch - 3) * IMHW; }
        else if (ch < 15) {
            const int t = ch - 6;                       // g_i * s_c
            p0 = g + (size_t)(t / 3) * IMHW;
            p1 = s + (size_t)(t % 3) * IMHW;
            mul = true;
        } else {                                        // rr rg rb gg gb bb
            const int t  = ch - 15;
            const int i0 = (t < 3) ? 0 : (t < 5 ? 1 : 2);
            const int i1 = (t == 0) ? 0 : ((t == 1 || t == 3) ? 1 : 2);
            p0 = g + (size_t)i0 * IMHW;
            p1 = g + (size_t)i1 * IMHW;
            mul = true;
        }
    } else {
        p0 = g + (size_t)ch * IMHW;
    }

    // Band fragments: depend only on lane and col0; computed once.
    v2f bb[12];
#pragma unroll
    for (int chunk = 0; chunk < 12; ++chunk) {
        const int kbase = chunk * 4 + half * 2;
        const int j0    = col0 - RAD + kbase;
        bb[chunk].x = (kbase     >= mn && kbase     <= mn + 2 * RAD && j0     >= 0 && j0     < IMW) ? 1.0f : 0.0f;
        bb[chunk].y = (kbase + 1 >= mn && kbase + 1 <= mn + 2 * RAD && j0 + 1 >= 0 && j0 + 1 < IMW) ? 1.0f : 0.0f;
    }

    const bool cInterior = (col0 >= RAD) && (col0 + 2 * RAD - 1 < IMW);   // wave-uniform
    const int  jbase     = col0 - RAD + half * 2;       // first column this lane touches
    const int  outcol    = col0 + mn;

    for (int rt = 0; rt < 8; ++rt) {
        const int row0  = (blockIdx.y * 8 + rt) * 16;
        const int rbase = (row0 + mn) * IMW;            // A row for this lane

        // Gather the 12 A fragments (loads only inside uniform branches).
        v2f afrag[12];
        if (cInterior) {
            const float* __restrict__ b0 = p0 + rbase + jbase;   // 8B aligned
            if (mul) {
                const float* __restrict__ b1 = p1 + rbase + jbase;
#pragma unroll
                for (int chunk = 0; chunk < 12; ++chunk)
                    afrag[chunk] = *(const v2f*)(b0 + chunk * 4) *
                                   *(const v2f*)(b1 + chunk * 4);
            } else {
#pragma unroll
                for (int chunk = 0; chunk < 12; ++chunk)
                    afrag[chunk] = *(const v2f*)(b0 + chunk * 4);
            }
        } else {                                         // edge strip: clamped scalar
            if (mul) {
#pragma unroll
                for (int chunk = 0; chunk < 12; ++chunk) {
                    const int c0 = clampi(jbase + chunk * 4,     0, IMW - 1);
                    const int c1 = clampi(jbase + chunk * 4 + 1, 0, IMW - 1);
                    afrag[chunk].x = p0[rbase + c0] * p1[rbase + c0];
                    afrag[chunk].y = p0[rbase + c1] * p1[rbase + c1];
                }
            } else {
#pragma unroll
                for (int chunk = 0; chunk < 12; ++chunk) {
                    const int c0 = clampi(jbase + chunk * 4,     0, IMW - 1);
                    const int c1 = clampi(jbase + chunk * 4 + 1, 0, IMW - 1);
                    afrag[chunk].x = p0[rbase + c0];
                    afrag[chunk].y = p0[rbase + c1];
                }
            }
        }

        // Single WMMA chain (not cloned by boundary handling).
        v8f acc = {};
#pragma unroll
        for (int chunk = 0; chunk < 12; ++chunk)
            acc = __builtin_amdgcn_wmma_f32_16x16x4_f32(false, afrag[chunk], false,
                                                        bb[chunk], (short)0, acc,
                                                        false, false);
#pragma unroll
        for (int v = 0; v < 8; ++v)
            out[ch * IMHW + (row0 + half * 8 + v) * IMW + outcol] = acc[v];
    }
}

// ---------------------------------------------------------------------------
// Vertical clamped box sum: out(16x16) = Band(16x48) x in(48x16).
// 12 B-fragments in a register ring; 4 new fragments per 16-row advance.
// Interior tiles use the lane-constant band; first/last tiles mask it.
// ---------------------------------------------------------------------------
__global__ void vbox_kernel(const float* __restrict__ in, float* __restrict__ out) {
    const int lane = threadIdx.x & 31;
    const int wave = threadIdx.x >> 5;
    const int half = lane >> 4;
    const int mn   = lane & 15;
    const int col  = (blockIdx.x * 8 + wave) * 16 + mn;
    const int ch   = blockIdx.z;
    const float* __restrict__ plane = in + (size_t)ch * IMHW;
    float* __restrict__ oplane      = out + (size_t)ch * IMHW;

    const bool topStrip = (blockIdx.y == 0);
    const bool botStrip = (blockIdx.y == gridDim.y - 1);

    // Interior band: Band[m][k] = 1 iff m <= k <= m+32 (lane-constant).
    v2f band_i[12];
#pragma unroll
    for (int chunk = 0; chunk < 12; ++chunk) {
        const int kbase = chunk * 4 + half * 2;
        band_i[chunk].x = (kbase     >= mn && kbase     <= mn + 2 * RAD) ? 1.0f : 0.0f;
        band_i[chunk].y = (kbase + 1 >= mn && kbase + 1 <= mn + 2 * RAD) ? 1.0f : 0.0f;
    }

    // Ring fill: F[c] holds rows (row0 - 16 + 4c + half*2 + {0,1}) at this column.
    int row0 = blockIdx.y * 128;
    v2f F[12];
    if (topStrip) {                                      // clamp above image
#pragma unroll
        for (int chunk = 0; chunk < 12; ++chunk) {
            const int r = row0 - RAD + chunk * 4 + half * 2;
            F[chunk].x = plane[clampi(r,     0, IMH - 1) * IMW + col];
            F[chunk].y = plane[clampi(r + 1, 0, IMH - 1) * IMW + col];
        }
    } else {                                             // direct, offsets fold
        const float* __restrict__ bp = plane + (row0 - RAD + half * 2) * IMW + col;
#pragma unroll
        for (int chunk = 0; chunk < 12; ++chunk) {
            F[chunk].x = bp[(chunk * 4)     * IMW];
            F[chunk].y = bp[(chunk * 4 + 1) * IMW];
        }
    }

    for (int rt = 0; rt < 8; ++rt, row0 += 16) {
        // Current band (mask only on the two boundary tiles; uniform test).
        v2f bcur[12];
        const bool interior = (row0 >= RAD) && (row0 + 2 * RAD - 1 < IMH);
        if (interior) {
#pragma unroll
            for (int c = 0; c < 12; ++c) bcur[c] = band_i[c];
        } else {
#pragma unroll
            for (int c = 0; c < 12; ++c) {
                const int r = row0 - RAD + c * 4 + half * 2;
                bcur[c].x = (band_i[c].x != 0.0f && r     >= 0 && r     < IMH) ? 1.0f : 0.0f;
                bcur[c].y = (band_i[c].y != 0.0f && r + 1 >= 0 && r + 1 < IMH) ? 1.0f : 0.0f;
            }
        }

        v8f acc = {};
#pragma unroll
        for (int chunk = 0; chunk < 12; ++chunk)
            acc = __builtin_amdgcn_wmma_f32_16x16x4_f32(false, bcur[chunk], false,
                                                        F[chunk], (short)0, acc,
                                                        false, false);
#pragma unroll
        for (int v = 0; v < 8; ++v)
            oplane[(row0 + half * 8 + v) * IMW + col] = acc[v];

        if (rt < 7) {                                    // reuse 32 rows, load 16 new
#pragma unroll
            for (int c = 0; c < 8; ++c) F[c] = F[c + 4];
            // New rows: row0+32 .. row0+47 at this column.
            if (botStrip) {
#pragma unroll
                for (int c = 0; c < 4; ++c) {
                    const int r = row0 + 2 * RAD + c * 4 + half * 2;
                    F[8 + c].x = plane[clampi(r,     0, IMH - 1) * IMW + col];
                    F[8 + c].y = plane[clampi(r + 1, 0, IMH - 1) * IMW + col];
                }
            } else {
                const float* __restrict__ bp = plane + (row0 + 2 * RAD + half * 2) * IMW + col;
#pragma unroll
                for (int c = 0; c < 4; ++c) {
                    F[8 + c].x = bp[(c * 4)     * IMW];
                    F[8 + c].y = bp[(c * 4 + 1) * IMW];
                }
            }
        }
    }
}

// ---------------------------------------------------------------------------
// Per-pixel 3x3 symmetric inverse + a/b computation. N analytic.
// S: 21 box-summed planes.  AB out: planes 0..8 = a[j*3+c], 9..11 = b[c].
// ---------------------------------------------------------------------------
__global__ void solve_kernel(const float* __restrict__ S, float* __restrict__ AB) {
    const int idx = blockIdx.x * blockDim.x + threadIdx.x;
    if (idx >= IMHW) return;
    const int x = idx & (IMW - 1);
    const int y = idx >> 11;
    const float nx   = (float)(min(x + RAD, IMW - 1) - max(x - RAD, 0) + 1);
    const float ny   = (float)(min(y + RAD, IMH - 1) - max(y - RAD, 0) + 1);
    const float invN = 1.0f / (nx * ny);

    float mI[3], mp[3], cov[3][3];
#pragma unroll
    for (int i = 0; i < 3; ++i) mI[i] = S[i * IMHW + idx] * invN;
#pragma unroll
    for (int c = 0; c < 3; ++c) mp[c] = S[(3 + c) * IMHW + idx] * invN;
#pragma unroll
    for (int i = 0; i < 3; ++i)
#pragma unroll
        for (int c = 0; c < 3; ++c)
            cov[i][c] = S[(6 + 3 * i + c) * IMHW + idx] * invN - mI[i] * mp[c];

    const float eps  = 1.0e-4f;
    const float v_rr = S[15 * IMHW + idx] * invN - mI[0] * mI[0] + eps;
    const float v_rg = S[16 * IMHW + idx] * invN - mI[0] * mI[1];
    const float v_rb = S[17 * IMHW + idx] * invN - mI[0] * mI[2];
    const float v_gg = S[18 * IMHW + idx] * invN - mI[1] * mI[1] + eps;
    const float v_gb = S[19 * IMHW + idx] * invN - mI[1] * mI[2];
    const float v_bb = S[20 * IMHW + idx] * invN - mI[2] * mI[2] + eps;

    const float det = v_rr * (v_gg * v_bb - v_gb * v_gb)
                    - v_rg * (v_rg * v_bb - v_gb * v_rb)
                    + v_rb * (v_rg * v_gb - v_gg * v_rb);
    const float id = 1.0f / det;

    float inv[3][3];
    inv[0][0] =  (v_gg * v_bb - v_gb * v_gb) * id;
    inv[0][1] = -(v_rg * v_bb - v_rb * v_gb) * id;
    inv[0][2] =  (v_rg * v_gb - v_rb * v_gg) * id;
    inv[1][1] =  (v_rr * v_bb - v_rb * v_rb) * id;
    inv[1][2] = -(v_rr * v_gb - v_rb * v_rg) * id;
    inv[2][2] =  (v_rr * v_gg - v_rg * v_rg) * id;
    inv[1][0] = inv[0][1]; inv[2][0] = inv[0][2]; inv[2][1] = inv[1][2];

    float a[3][3];
#pragma unroll
    for (int j = 0; j < 3; ++j)
#pragma unroll
        for (int c = 0; c < 3; ++c) {
            float acc = 0.0f;
#pragma unroll
            for (int i = 0; i < 3; ++i) acc += inv[j][i] * cov[i][c];
            a[j][c] = acc;
            AB[(3 * j + c) * IMHW + idx] = acc;
        }
#pragma unroll
    for (int c = 0; c < 3; ++c) {
        float bb = mp[c];
#pragma unroll
        for (int j = 0; j < 3; ++j) bb -= a[j][c] * mI[j];
        AB[(9 + c) * IMHW + idx] = bb;
    }
}

// q[c] = (box(a)[j,c]·guide[j] + box(b)[c]) / N
__global__ void final_kernel(const float* __restrict__ M, const float* __restrict__ g,
                             float* __restrict__ q) {
    const int idx = blockIdx.x * blockDim.x + threadIdx.x;
    if (idx >= IMHW) return;
    const int x = idx & (IMW - 1);
    const int y = idx >> 11;
    const float invN = 1.0f / ((float)(min(x + RAD, IMW - 1) - max(x - RAD, 0) + 1) *
                               (float)(min(y + RAD, IMH - 1) - max(y - RAD, 0) + 1));
    const float gr = g[idx], gg = g[IMHW + idx], gb = g[2 * IMHW + idx];
#pragma unroll
    for (int c = 0; c < 3; ++c) {
        float acc = M[(9 + c) * IMHW + idx];
        acc += M[(0 + c) * IMHW + idx] * gr
             + M[(3 + c) * IMHW + idx] * gg
             + M[(6 + c) * IMHW + idx] * gb;
        q[c * IMHW + idx] = acc * invN;
    }
}

extern "C" void kernel_launch(void* const* d_in, const int* in_sizes, int n_in,
                              void* d_out, int out_size, void* d_ws, size_t ws_size,
                              hipStream_t stream) {
    (void)in_sizes; (void)n_in; (void)out_size; (void)ws_size;
    const float* guide = (const float*)d_in[0];
    const float* src   = (const float*)d_in[1];
    float* out  = (float*)d_out;
    float* bufA = (float*)d_ws;                          // 21 planes (336 MB)
    float* bufB = bufA + (size_t)NCH1 * IMHW;            // 21 planes (336 MB)

    const dim3 blk(256);                                 // 8 waves, 1 strip each
    const dim3 g21(IMW / 16 / 8, IMH / 16 / 8, NCH1);    // (16, 16, 21)
    const dim3 g12(IMW / 16 / 8, IMH / 16 / 8, NCH2);    // (16, 16, 12)
    const int  pThreads = 256;
    const int  pBlocks  = IMHW / pThreads;

    hbox_kernel<0><<<g21, blk, 0, stream>>>(guide, src, bufA);   // products + h-box
    vbox_kernel   <<<g21, blk, 0, stream>>>(bufA, bufB);         // v-box -> stage-1 sums
    solve_kernel  <<<pBlocks, pThreads, 0, stream>>>(bufB, bufA);// 3x3 solve -> a,b
    hbox_kernel<1><<<g12, blk, 0, stream>>>(bufA, nullptr, bufB);// h-box a,b
    vbox_kernel   <<<g12, blk, 0, stream>>>(bufB, bufA);         // v-box a,b
    final_kernel  <<<pBlocks, pThreads, 0, stream>>>(bufA, guide, out);
}